// GAT_with_Residual_26628797235678
// MI455X (gfx1250) — compile-verified
//
#include <hip/hip_runtime.h>
#include <hip/hip_bf16.h>
#include <math.h>
#include <stdint.h>

// ---------------- constants from the reference ----------------
#define NN   50000
#define EE   800000
#define INC  128
#define HID  32
#define HEADS 4
#define OUTC 64
#define NEG_SLOPE 0.2f

typedef __attribute__((ext_vector_type(2))) float v2f;
typedef __attribute__((ext_vector_type(8))) float v8f;

#if __has_builtin(__builtin_amdgcn_global_load_async_to_lds_b128)
#define HAVE_ASYNC_LDS 1
typedef int v4i __attribute__((vector_size(16)));
typedef __attribute__((address_space(1))) v4i gv4i_t;  // global (== __device__)
typedef __attribute__((address_space(3))) v4i lv4i_t;  // LDS    (== __shared__)
__device__ inline void async_b128(const void* g, void* l) {
  gv4i_t* gp = reinterpret_cast<gv4i_t*>(reinterpret_cast<uintptr_t>(g));
  lv4i_t* lp = reinterpret_cast<lv4i_t*>(reinterpret_cast<uintptr_t>(l));
  __builtin_amdgcn_global_load_async_to_lds_b128(gp, lp, 0, 0);
}
#else
#define HAVE_ASYNC_LDS 0
#endif

__device__ inline void wait_asynccnt0() {
#if __has_builtin(__builtin_amdgcn_s_wait_asynccnt)
  __builtin_amdgcn_s_wait_asynccnt(0);
#else
  asm volatile("s_wait_asynccnt 0x0" ::: "memory");
#endif
}

// =====================================================================
// WMMA GEMM:  Out[M,OUTCOLS] = A[M,K] @ W[K,OUTCOLS] (+bias) (+relu)
// One wave computes a 16-row x OUTCOLS tile with V_WMMA_F32_16X16X4_F32.
// Both W and the block's 128-row A tile are staged into LDS via async
// global->LDS DMA (ASYNCcnt), so the inner loop is pure ds_load + wmma
// (DScnt-pipelined; no global-load stalls).
// A rows padded to K+4 dwords: 16B-aligned b128 stores, and fragment
// ds_load_b64 reads hit banks {4r,4r+1}/{4r+2,4r+3} -> conflict-free.
//
// f32 WMMA fragment layouts (CDNA5 ISA 7.12.2):
//  A 16x4 : lane l -> row = l&15 ; v0 = A[row, k + 2*(l>>4)], v1 = +1
//  B 4x16 : lane l -> col = l&15 ; v0 = B[k + 2*(l>>4), col], v1 = +1
//  C/D    : vgpr r, lane l -> row = r + 8*(l>>4), col = l&15
// =====================================================================
template <int K, int OUTCOLS, bool RELU>
__global__ __launch_bounds__(256) void wmma_gemm(
    const float* __restrict__ A, const float* __restrict__ W,
    const float* __restrict__ bias, float* __restrict__ Out, int M) {
  constexpr int NT = OUTCOLS / 16;
  constexpr int WBYTES = K * OUTCOLS * 4;
  constexpr int KC = K / 4;        // 16B chunks per A row
  constexpr int APAD = K + 4;      // padded A row stride (dwords)
  __shared__ float sW[K * OUTCOLS];
  __shared__ float sA[128 * APAD];

  const int rowBase0 = blockIdx.x * 128;
#if HAVE_ASYNC_LDS
  {
    const char* gw = (const char*)W;
    char* lw = (char*)sW;
    for (int i = threadIdx.x * 16; i < WBYTES; i += 256 * 16)
      async_b128(gw + i, lw + i);
    for (int i = threadIdx.x; i < 128 * KC; i += 256) {
      int r = i / KC, c = i - (i / KC) * KC;
      int rr = rowBase0 + r;
      if (rr > M - 1) rr = M - 1;  // clamp: no OOB reads on partial block
      async_b128(A + (size_t)rr * K + c * 4, sA + r * APAD + c * 4);
    }
    wait_asynccnt0();
  }
#else
  for (int i = threadIdx.x; i < K * OUTCOLS; i += 256) sW[i] = W[i];
  for (int i = threadIdx.x; i < 128 * KC; i += 256) {
    int r = i / KC, c = i - (i / KC) * KC;
    int rr = rowBase0 + r;
    if (rr > M - 1) rr = M - 1;
    *(float4*)(sA + r * APAD + c * 4) =
        *(const float4*)(A + (size_t)rr * K + c * 4);
  }
#endif
  __syncthreads();

  const int wave = threadIdx.x >> 5;
  const int lane = threadIdx.x & 31;
  const int rowBase = rowBase0 + wave * 16;
  if (rowBase >= M) return;              // wave-uniform: EXEC stays all-ones

  const int r15 = lane & 15;
  const int khalf = (lane >> 4) * 2;     // 0 or 2
  const int rowHi = (lane >> 4) * 8;     // 0 or 8

  v8f acc[NT];
#pragma unroll
  for (int t = 0; t < NT; ++t) acc[t] = (v8f){0, 0, 0, 0, 0, 0, 0, 0};

  const float* aL = sA + (wave * 16 + r15) * APAD;  // lane's A row in LDS
  for (int k = 0; k < K; k += 4) {
    v2f a;
    a.x = aL[k + khalf];
    a.y = aL[k + khalf + 1];
#pragma unroll
    for (int t = 0; t < NT; ++t) {
      v2f b;
      b.x = sW[(k + khalf) * OUTCOLS + t * 16 + r15];
      b.y = sW[(k + khalf + 1) * OUTCOLS + t * 16 + r15];
      acc[t] = __builtin_amdgcn_wmma_f32_16x16x4_f32(
          false, a, false, b, (short)0, acc[t], false, false);
    }
  }

#pragma unroll
  for (int t = 0; t < NT; ++t) {
    const float bv = bias ? bias[t * 16 + r15] : 0.0f;
#pragma unroll
    for (int r = 0; r < 8; ++r) {
      float v = acc[t][r] + bv;
      if (RELU) v = v > 0.0f ? v : 0.0f;
      Out[(size_t)(rowBase + r + rowHi) * OUTCOLS + t * 16 + r15] = v;
    }
  }
}

// ---------------- attention coefficients: per (node, head) dot ----------------
template <int H, int C>
__global__ void attn_coeff(const float* __restrict__ h,
                           const float* __restrict__ asrc,
                           const float* __restrict__ adst,
                           float* __restrict__ outs, float* __restrict__ outd,
                           int n) {
  int t = blockIdx.x * blockDim.x + threadIdx.x;
  if (t >= n * H) return;
  int node = t / H, hd = t % H;
  const float* hp = h + (size_t)node * (H * C) + hd * C;
  const float* av = asrc + hd * C;
  const float* bv = adst + hd * C;
  float ss = 0.f, dd = 0.f;
#pragma unroll 8
  for (int c = 0; c < C; ++c) {
    float v = hp[c];
    ss += v * av[c];
    dd += v * bv[c];
  }
  outs[t] = ss;
  outd[t] = dd;
}

// ---------------- fill ----------------
__global__ void fillk(float* __restrict__ p, float v, int n) {
  int t = blockIdx.x * blockDim.x + threadIdx.x;
  if (t < n) p[t] = v;
}

// ---------------- float atomic max via int/uint atomics ----------------
__device__ inline void atomicMaxFloat(float* addr, float v) {
  if (v >= 0.0f)
    atomicMax((int*)addr, __float_as_int(v));
  else
    atomicMin((unsigned int*)addr, __float_as_uint(v));
}

__device__ inline void edge_sd(const int* __restrict__ ei, int idx, int& s,
                               int& d) {
  if (idx < EE) {
    s = ei[idx];
    d = ei[EE + idx];
  } else {
    s = d = idx - EE;  // self-loops appended after the edge list
  }
}

// ---------------- pass A: e = leaky_relu(as[src]+ad[dst]); segment max ----------------
template <int H>
__global__ void edge_max_k(const int* __restrict__ ei,
                           const float* __restrict__ as,
                           const float* __restrict__ ad,
                           float* __restrict__ ebuf, float* __restrict__ mm) {
  int t = blockIdx.x * blockDim.x + threadIdx.x;
  const int total = (EE + NN) * H;
  if (t >= total) return;
  int idx = t / H, h = t - idx * H;
  int s, d;
  edge_sd(ei, idx, s, d);
  float e = as[s * H + h] + ad[d * H + h];
  e = e > 0.0f ? e : NEG_SLOPE * e;
  ebuf[t] = e;
  atomicMaxFloat(mm + d * H + h, e);
}

// ---------------- pass B: ee = exp(e - m[dst]); segment sum ----------------
template <int H>
__global__ void edge_exp_k(const int* __restrict__ ei, float* __restrict__ ebuf,
                           const float* __restrict__ mm,
                           float* __restrict__ zz) {
  int t = blockIdx.x * blockDim.x + threadIdx.x;
  const int total = (EE + NN) * H;
  if (t >= total) return;
  int idx = t / H, h = t - idx * H;
  int s, d;
  edge_sd(ei, idx, s, d);
  float e = __expf(ebuf[t] - mm[d * H + h]);
  ebuf[t] = e;
  atomicAdd(zz + d * H + h, e);
}

// ---------------- pass C: out[dst] += (ee/z[dst]) * h[src]  (block per edge) ----------------
template <int H, int C>
__global__ void edge_agg_k(const int* __restrict__ ei,
                           const float* __restrict__ ebuf,
                           const float* __restrict__ zz,
                           const float* __restrict__ hpre,
                           float* __restrict__ out) {
  int idx = blockIdx.x;            // one block per edge (incl. self loops)
  int tid = threadIdx.x;           // 0 .. H*C-1
  int h = tid / C;
  int s, d;
  edge_sd(ei, idx, s, d);
  float z = fmaxf(zz[d * H + h], 1e-16f);
  float alpha = ebuf[(size_t)idx * H + h] / z;
  atomicAdd(&out[(size_t)d * (H * C) + tid],
            alpha * hpre[(size_t)s * (H * C) + tid]);
}

// ---------------- y = elu(y + bias[col]), in place ----------------
template <int COLS>
__global__ void elu_bias_k(float* __restrict__ p, const float* __restrict__ b,
                           int total) {
  int t = blockIdx.x * blockDim.x + threadIdx.x;
  if (t >= total) return;
  float v = p[t] + b[t & (COLS - 1)];
  p[t] = v > 0.0f ? v : expm1f(v);
}

// ---------------- hfin = elu(agg2 + b2) + res ----------------
__global__ void combine_k(const float* __restrict__ agg,
                          const float* __restrict__ b2,
                          const float* __restrict__ res,
                          float* __restrict__ out, int total) {
  int t = blockIdx.x * blockDim.x + threadIdx.x;
  if (t >= total) return;
  float v = agg[t] + b2[t & 63];
  v = v > 0.0f ? v : expm1f(v);
  out[t] = v + res[t];
}

// ---------------- out[N,2] = c1 @ Wc2 + bc2 ----------------
__global__ void final_k(const float* __restrict__ c1,
                        const float* __restrict__ Wc2,
                        const float* __restrict__ bc2, float* __restrict__ out,
                        int n) {
  int t = blockIdx.x * blockDim.x + threadIdx.x;
  if (t >= n) return;
  const float* r = c1 + (size_t)t * 64;
  float o0 = bc2[0], o1 = bc2[1];
#pragma unroll 8
  for (int k = 0; k < 64; ++k) {
    float v = r[k];
    o0 += v * Wc2[k * 2 + 0];
    o1 += v * Wc2[k * 2 + 1];
  }
  out[t * 2 + 0] = o0;
  out[t * 2 + 1] = o1;
}

static inline int cdiv(int a, int b) { return (a + b - 1) / b; }

extern "C" void kernel_launch(void* const* d_in, const int* in_sizes, int n_in,
                              void* d_out, int out_size, void* d_ws,
                              size_t ws_size, hipStream_t stream) {
  const float* x      = (const float*)d_in[0];
  const int*   ei     = (const int*)d_in[1];
  const float* W1     = (const float*)d_in[2];
  const float* a_src1 = (const float*)d_in[3];
  const float* a_dst1 = (const float*)d_in[4];
  const float* b1     = (const float*)d_in[5];
  const float* W2     = (const float*)d_in[6];
  const float* a_src2 = (const float*)d_in[7];
  const float* a_dst2 = (const float*)d_in[8];
  const float* b2     = (const float*)d_in[9];
  const float* Wres   = (const float*)d_in[10];
  const float* bres   = (const float*)d_in[11];
  const float* Wc1    = (const float*)d_in[12];
  const float* bc1    = (const float*)d_in[13];
  const float* Wc2    = (const float*)d_in[14];
  const float* bc2    = (const float*)d_in[15];
  float* out = (float*)d_out;

  // ---- workspace layout (floats), with overlays ----
  float* ws    = (float*)d_ws;
  float* h1pre = ws;                           // N*128
  float* h1out = ws + (size_t)NN * 128;        // N*128
  float* as    = h1out + (size_t)NN * 128;     // N*4
  float* ad    = as + NN * 4;                  // N*4
  float* mm    = ad + NN * 4;                  // N*4
  float* zz    = mm + NN * 4;                  // N*4
  float* ebuf  = zz + NN * 4;                  // (E+N)*4
  // overlays (lifetimes are disjoint):
  float* h2pre = h1pre;                        // N*64  (h1pre dead)
  float* agg2  = h1pre + (size_t)NN * 64;      // N*64
  float* res   = h1out;                        // N*64  (h1out dead)
  float* hfin  = h1out + (size_t)NN * 64;      // N*64
  float* c1    = h1pre;                        // N*64  (h2pre dead)

  const float NEGINF = -INFINITY;
  const int EL = EE + NN;  // edges incl. self loops

  // ================= layer 1 =================
  wmma_gemm<128, 128, false><<<cdiv(NN, 128), 256, 0, stream>>>(
      x, W1, nullptr, h1pre, NN);
  attn_coeff<HEADS, HID><<<cdiv(NN * HEADS, 256), 256, 0, stream>>>(
      h1pre, a_src1, a_dst1, as, ad, NN);
  fillk<<<cdiv(NN * HEADS, 256), 256, 0, stream>>>(mm, NEGINF, NN * HEADS);
  fillk<<<cdiv(NN * HEADS, 256), 256, 0, stream>>>(zz, 0.0f, NN * HEADS);
  fillk<<<cdiv(NN * 128, 256), 256, 0, stream>>>(h1out, 0.0f, NN * 128);
  edge_max_k<HEADS><<<cdiv(EL * HEADS, 256), 256, 0, stream>>>(ei, as, ad,
                                                               ebuf, mm);
  edge_exp_k<HEADS><<<cdiv(EL * HEADS, 256), 256, 0, stream>>>(ei, ebuf, mm,
                                                               zz);
  edge_agg_k<HEADS, HID><<<EL, HEADS * HID, 0, stream>>>(ei, ebuf, zz, h1pre,
                                                         h1out);
  elu_bias_k<128><<<cdiv(NN * 128, 256), 256, 0, stream>>>(h1out, b1, NN * 128);

  // ================= layer 2 =================
  wmma_gemm<128, 64, false><<<cdiv(NN, 128), 256, 0, stream>>>(
      h1out, W2, nullptr, h2pre, NN);
  attn_coeff<1, 64><<<cdiv(NN, 256), 256, 0, stream>>>(h2pre, a_src2, a_dst2,
                                                       as, ad, NN);
  fillk<<<cdiv(NN, 256), 256, 0, stream>>>(mm, NEGINF, NN);
  fillk<<<cdiv(NN, 256), 256, 0, stream>>>(zz, 0.0f, NN);
  fillk<<<cdiv(NN * 64, 256), 256, 0, stream>>>(agg2, 0.0f, NN * 64);
  edge_max_k<1><<<cdiv(EL, 256), 256, 0, stream>>>(ei, as, ad, ebuf, mm);
  edge_exp_k<1><<<cdiv(EL, 256), 256, 0, stream>>>(ei, ebuf, mm, zz);
  edge_agg_k<1, 64><<<EL, 64, 0, stream>>>(ei, ebuf, zz, h2pre, agg2);

  // ============ residual + combine ============
  wmma_gemm<128, 64, false><<<cdiv(NN, 128), 256, 0, stream>>>(
      x, Wres, bres, res, NN);
  combine_k<<<cdiv(NN * 64, 256), 256, 0, stream>>>(agg2, b2, res, hfin,
                                                    NN * 64);

  // ============ classifier head ============
  wmma_gemm<64, 64, true><<<cdiv(NN, 128), 256, 0, stream>>>(hfin, Wc1, bc1,
                                                             c1, NN);
  final_k<<<cdiv(NN, 256), 256, 0, stream>>>(c1, Wc2, bc2, out, NN);
}